// LSTMEncoder_83373905150639
// MI455X (gfx1250) — compile-verified
//
#include <hip/hip_runtime.h>

typedef __attribute__((ext_vector_type(2))) float v2f;
typedef __attribute__((ext_vector_type(8))) float v8f;

namespace {
constexpr int kB = 4096;
constexpr int kT = 200;
constexpr int kI = 32;      // input dim
constexpr int kH = 64;      // hidden dim
constexpr int kG = 4 * kH;  // 256 gate columns (i,f,g,o)
constexpr int kP = kH / 2;  // 32 projection dim
constexpr int kRows = 16;   // batch rows per workgroup (one WMMA M tile)
constexpr int kThreads = 128;  // 4 waves of 32
}  // namespace

__device__ __forceinline__ v8f wmma_f32(v2f a, v2f b, v8f c) {
  // D = A(16x4, f32) * B(4x16, f32) + C(16x16, f32)
  return __builtin_amdgcn_wmma_f32_16x16x4_f32(false, a, false, b, (short)0, c,
                                               false, false);
}

__device__ __forceinline__ float fast_sigmoid(float x) {
  // 1 / (1 + 2^(-x*log2(e)))
  return __builtin_amdgcn_rcpf(
      1.0f + __builtin_amdgcn_exp2f(-1.4426950408889634f * x));
}
__device__ __forceinline__ float fast_tanh(float x) {
  // tanh(x) = 1 - 2/(exp(2x)+1); exp(2x) = 2^(2x*log2(e))
  return 1.0f - 2.0f * __builtin_amdgcn_rcpf(
                           1.0f + __builtin_amdgcn_exp2f(
                                      2.8853900817779268f * x));
}

__global__ __launch_bounds__(kThreads, 1) void lstm_fused_kernel(
    const float* __restrict__ x,       // [B, T, I]
    const float* __restrict__ W_ih,    // [4H, I]
    const float* __restrict__ W_hh,    // [4H, H]
    const float* __restrict__ b_ih,    // [4H]
    const float* __restrict__ b_hh,    // [4H]
    const float* __restrict__ W_proj,  // [P, H]
    const float* __restrict__ b_proj,  // [P]
    float* __restrict__ out) {         // [B, P]
  __shared__ __align__(16) float hbuf[2][kRows * kH];  // ping-pong h, 8 KB
  __shared__ __align__(16) float wih_s[kG * kI];       // W_ih copy, 32 KB

  const int tid = threadIdx.x;
  const int lane = tid & 31;
  const int j = tid >> 5;       // wave id == 16-wide column tile within a gate
  const int half = lane >> 4;   // 0/1 (lane group)
  const int l16 = lane & 15;
  const int rowbase = blockIdx.x * kRows;

  // Stage W_ih into LDS (coalesced float4 copy).
  for (int idx = tid; idx < (kG * kI) / 4; idx += kThreads) {
    ((float4*)wih_s)[idx] = ((const float4*)W_ih)[idx];
  }
  // Zero initial h.
  for (int k2 = tid; k2 < kRows * kH; k2 += kThreads) hbuf[0][k2] = 0.0f;

  // Fused bias per gate for this lane's output column (C/D layout: N = l16).
  float bias[4];
  int wi_off[4];
#pragma unroll
  for (int g = 0; g < 4; ++g) {
    const int col = (g * 4 + j) * 16 + l16;  // absolute gate column
    bias[g] = b_ih[col] + b_hh[col];
    wi_off[g] = col * kI + 2 * half;  // B-operand lane offset into wih_s
  }

  // Register-resident W_hh^T B-operand tiles: lane holds
  // (W_hh[col][k], W_hh[col][k+1]) with col = N = l16 of tile, k = 4*kk+2*half.
  v2f whh[4][16];
#pragma unroll
  for (int g = 0; g < 4; ++g) {
    const float* wh =
        W_hh + (size_t)((g * 4 + j) * 16 + l16) * kH + 2 * half;
#pragma unroll
    for (int kk = 0; kk < 16; ++kk) whh[g][kk] = *(const v2f*)(wh + 4 * kk);
  }

  // A-operand addressing: lane row = l16, K pair = 4*kk + 2*half.
  const float* xrow = x + (size_t)(rowbase + l16) * kT * kI + 2 * half;
  const int a_off = l16 * kH + 2 * half;  // into hbuf

  v8f c = {};  // cell state tile (C/D layout): row m = e + 8*half, col = l16
  __syncthreads();

#pragma unroll 1
  for (int t = 0; t < kT; ++t) {
    const float* hcur = hbuf[t & 1];
    float* hnxt = hbuf[(t & 1) ^ 1];

    // Issue this step's x fetch first (A-operand chunks into registers) ...
    const float* xp = xrow + t * kI;
    v2f xa[8];
#pragma unroll
    for (int kk = 0; kk < 8; ++kk) xa[kk] = *(const v2f*)(xp + 4 * kk);
    // ... and warm the WGP cache for next step's x row (no counter cost).
    if (t + 1 < kT) __builtin_prefetch(xp + kI, 0, 0);

    v8f acc0 = {}, acc1 = {}, acc2 = {}, acc3 = {};

    // h contribution first: operands come from LDS (ready right after the
    // barrier), hiding the global-memory latency of the x fetch behind
    // 64 WMMAs of recurrent GEMM work.
#pragma unroll
    for (int kk = 0; kk < 16; ++kk) {
      const v2f a = *(const v2f*)(hcur + a_off + 4 * kk);
      acc0 = wmma_f32(a, whh[0][kk], acc0);
      acc1 = wmma_f32(a, whh[1][kk], acc1);
      acc2 = wmma_f32(a, whh[2][kk], acc2);
      acc3 = wmma_f32(a, whh[3][kk], acc3);
    }
    // x contribution: K = 32 (8 chained K=4 WMMAs per gate tile)
#pragma unroll
    for (int kk = 0; kk < 8; ++kk) {
      acc0 = wmma_f32(xa[kk], *(const v2f*)(wih_s + wi_off[0] + 4 * kk), acc0);
      acc1 = wmma_f32(xa[kk], *(const v2f*)(wih_s + wi_off[1] + 4 * kk), acc1);
      acc2 = wmma_f32(xa[kk], *(const v2f*)(wih_s + wi_off[2] + 4 * kk), acc2);
      acc3 = wmma_f32(xa[kk], *(const v2f*)(wih_s + wi_off[3] + 4 * kk), acc3);
    }

    // Elementwise LSTM cell on the 16x16 tile (8 elems/lane, i/f/g/o aligned).
#pragma unroll
    for (int e = 0; e < 8; ++e) {
      const float iv = fast_sigmoid(acc0[e] + bias[0]);
      const float fv = fast_sigmoid(acc1[e] + bias[1]);
      const float gv = fast_tanh(acc2[e] + bias[2]);
      const float ov = fast_sigmoid(acc3[e] + bias[3]);
      const float cv = fv * c[e] + iv * gv;
      c[e] = cv;
      // C/D layout: row m = e + 8*half, col n = j*16 + l16
      hnxt[(e + 8 * half) * kH + j * 16 + l16] = ov * fast_tanh(cv);
    }
    __syncthreads();
  }

  // Projection: out = relu(h_last @ W_proj^T + b_proj), P = 32 (2 N tiles).
  if (j < 2) {
    const float* hlast = hbuf[kT & 1];
    const float* wp = W_proj + (size_t)(j * 16 + l16) * kH + 2 * half;
    v8f acc = {};
#pragma unroll
    for (int kk = 0; kk < 16; ++kk) {
      const v2f a = *(const v2f*)(hlast + a_off + 4 * kk);
      const v2f b = *(const v2f*)(wp + 4 * kk);
      acc = wmma_f32(a, b, acc);
    }
    const int col = j * 16 + l16;
    const float bp = b_proj[col];
#pragma unroll
    for (int e = 0; e < 8; ++e) {
      float v = acc[e] + bp;
      v = v > 0.0f ? v : 0.0f;
      out[(size_t)(rowbase + e + 8 * half) * kP + col] = v;
    }
  }
}

extern "C" void kernel_launch(void* const* d_in, const int* in_sizes, int n_in,
                              void* d_out, int out_size, void* d_ws,
                              size_t ws_size, hipStream_t stream) {
  (void)in_sizes; (void)n_in; (void)out_size; (void)d_ws; (void)ws_size;
  const float* x = (const float*)d_in[0];
  const float* W_ih = (const float*)d_in[1];
  const float* W_hh = (const float*)d_in[2];
  const float* b_ih = (const float*)d_in[3];
  const float* b_hh = (const float*)d_in[4];
  const float* W_proj = (const float*)d_in[5];
  const float* b_proj = (const float*)d_in[6];
  float* out = (float*)d_out;

  dim3 grid(kB / kRows);  // 256 workgroups of 4 waves
  dim3 block(kThreads);
  lstm_fused_kernel<<<grid, block, 0, stream>>>(x, W_ih, W_hh, b_ih, b_hh,
                                                W_proj, b_proj, out);
}